// MaxPooling_34531537060558
// MI455X (gfx1250) — compile-verified
//
#include <hip/hip_runtime.h>

// Segment max (jax.ops.segment_max) for x:[N,128] f32, batch:[N] sorted int,
// out:[B,128] f32 with -inf for empty segments.
//
// Memory-bound streaming reduction: 512MB of x read once -> ~22us floor at
// 23.3 TB/s HBM. No matmul structure -> WMMA inapplicable. CDNA5 paths used:
// b128 non-temporal global loads, v_max_num_f32, native
// GLOBAL_ATOMIC_MAX_NUM_F32 at device scope, wave32 ballot boundary masks.

typedef float f4 __attribute__((ext_vector_type(4)));

#define DIMS 128          // feature dim (reference D)
#define ROWS_PER_WAVE 64  // contiguous rows owned by each wave32

// ---------------------------------------------------------------------------
// Kernel 1: fill output with -inf (empty segments must stay -inf).
// ---------------------------------------------------------------------------
__global__ __launch_bounds__(256) void seg_fill_neginf(f4* __restrict__ out, int n4) {
  int i = blockIdx.x * blockDim.x + threadIdx.x;
  if (i < n4) {
    const float ninf = -__builtin_inff();
    f4 v = {ninf, ninf, ninf, ninf};
    out[i] = v;
  }
}

// Native CDNA5 float-max atomic, device scope (L2-coherent across WGPs).
__device__ __forceinline__ void atomic_fmax_dev(float* p, float v) {
  asm volatile("global_atomic_max_num_f32 %0, %1, off scope:SCOPE_DEV"
               :
               : "v"(p), "v"(v)
               : "memory");
}

// Flush one lane's float4 accumulator (columns 4*lane .. 4*lane+3) of segment
// `seg` into the output with 4 device-scope float-max atomics.
__device__ __forceinline__ void flush_acc(float* __restrict__ out, int seg,
                                          int lane, const f4& acc) {
  float* o = out + (size_t)seg * DIMS + lane * 4;
  atomic_fmax_dev(o + 0, acc.x);
  atomic_fmax_dev(o + 1, acc.y);
  atomic_fmax_dev(o + 2, acc.z);
  atomic_fmax_dev(o + 3, acc.w);
}

// ---------------------------------------------------------------------------
// Kernel 2: each wave32 reduces ROWS_PER_WAVE consecutive rows.
// Lane l keeps a running float4 max of columns [4l,4l+3]. Segment boundaries
// inside the chunk are precomputed ONCE as a 64-bit ballot mask, so the
// streaming inner loop is pure load_b128(NT) + 4x v_max_num_f32 — no per-row
// shuffles, no per-row branches.
// ---------------------------------------------------------------------------
__global__ __launch_bounds__(256) void seg_max_kernel(const f4* __restrict__ x,
                                                      const int* __restrict__ batch,
                                                      float* __restrict__ out,
                                                      int n_rows) {
  const int lane = threadIdx.x & 31;
  const long long wave =
      (long long)blockIdx.x * (blockDim.x >> 5) + (threadIdx.x >> 5);
  const long long r0 = wave * (long long)ROWS_PER_WAVE;
  if (r0 >= n_rows) return;
  long long r1 = r0 + ROWS_PER_WAVE;
  if (r1 > n_rows) r1 = n_rows;
  const int nr = (int)(r1 - r0);

  // --- Preload this chunk's 64 segment ids (2 coalesced loads). ------------
  const long long last = (long long)n_rows - 1;
  long long i0 = r0 + lane;       if (i0 > last) i0 = last;
  long long i1 = r0 + 32 + lane;  if (i1 > last) i1 = last;
  const int segs0 = batch[i0];
  const int segs1 = batch[i1];

  // --- Build 64-bit boundary mask: bit k set iff id(row k) != id(row k-1). -
  const int up0   = __shfl_up(segs0, 1);            // lane0 -> own value
  const int last0 = __shfl(segs0, 31);
  const int up1   = (lane == 0) ? last0 : __shfl_up(segs1, 1);
  unsigned long long m0 = __ballot((lane != 0) && (segs0 != up0));
  unsigned long long m1 = __ballot(segs1 != up1);
  unsigned long long mask = (m0 & 0xffffffffull) | (m1 << 32);
  if (nr < 64) mask &= ((1ull << nr) - 1ull);       // ignore tail beyond chunk

  const float ninf = -__builtin_inff();
  f4 acc = {ninf, ninf, ninf, ninf};
  int cur = __shfl(segs0, 0);

  const f4* __restrict__ xrow = x + (size_t)r0 * (DIMS / 4) + lane;

  // --- Iterate over boundary-delimited runs. -------------------------------
  int s = 0;
  for (;;) {
    const int e = mask ? (int)__builtin_ctzll(mask) : nr;

    // Streaming run [s, e): pure b128-NT load + 4x max per row.
#pragma unroll 4
    for (int k = s; k < e; ++k) {
      f4 v = __builtin_nontemporal_load(xrow + (size_t)k * (DIMS / 4));
      acc.x = fmaxf(acc.x, v.x);
      acc.y = fmaxf(acc.y, v.y);
      acc.z = fmaxf(acc.z, v.z);
      acc.w = fmaxf(acc.w, v.w);
    }

    // Segment may extend beyond this chunk -> merge via atomic max.
    flush_acc(out, cur, lane, acc);

    if (e >= nr) break;
    cur = __shfl((e < 32) ? segs0 : segs1, e & 31);  // id of the new run
    acc.x = ninf; acc.y = ninf; acc.z = ninf; acc.w = ninf;
    mask &= mask - 1;                                // consume this boundary
    s = e;
  }
}

// ---------------------------------------------------------------------------
// Host launcher. Inputs per harness ABI: d_in[0]=x (float32, N*128),
// d_in[1]=batch (integer -> const int*), d_in[2]=dim_size (scalar, unused:
// B is recovered from out_size). Output: d_out = [B,128] float32.
// ---------------------------------------------------------------------------
extern "C" void kernel_launch(void* const* d_in, const int* in_sizes, int n_in,
                              void* d_out, int out_size, void* d_ws, size_t ws_size,
                              hipStream_t stream) {
  (void)n_in; (void)d_ws; (void)ws_size;
  const f4* x = (const f4*)d_in[0];
  const int* batch = (const int*)d_in[1];
  float* out = (float*)d_out;

  const int n_rows = in_sizes[0] / DIMS;

  // Phase 1: out = -inf everywhere (stream-ordered before phase 2).
  const int n4 = out_size / 4;
  const int fill_blocks = (n4 + 255) / 256;
  seg_fill_neginf<<<fill_blocks, 256, 0, stream>>>((f4*)d_out, n4);

  // Phase 2: one wave32 per 64-row chunk; 8 waves per 256-thread block.
  const long long nwaves =
      ((long long)n_rows + ROWS_PER_WAVE - 1) / ROWS_PER_WAVE;
  const int nblocks = (int)((nwaves + 7) / 8);
  seg_max_kernel<<<nblocks, 256, 0, stream>>>(x, batch, out, n_rows);
}